// MixHopLayer_69234872811809
// MI455X (gfx1250) — compile-verified
//
#include <hip/hip_runtime.h>
#include <hip/hip_bf16.h>

typedef float v2f __attribute__((ext_vector_type(2)));
typedef float v8f __attribute__((ext_vector_type(8)));

#define IN_C  128
#define OUT_C 128

// Guaranteed-native f32 atomic add (global_atomic_add_f32, agent scope in L2).
__device__ __forceinline__ void atomic_fadd_native(float* p, float v) {
#if defined(__HIP_DEVICE_COMPILE__)
    unsafeAtomicAdd(p, v);   // lowers to global_atomic_add_f32 on AMDGCN
#else
    __hip_atomic_fetch_add(p, v, __ATOMIC_RELAXED, __HIP_MEMORY_SCOPE_AGENT);
#endif
}

// ---------------------------------------------------------------------------
// Zero-fill (vectorized)
// ---------------------------------------------------------------------------
__global__ __launch_bounds__(256) void mh_zero4(float4* __restrict__ p, long n4) {
    long i = (long)blockIdx.x * blockDim.x + threadIdx.x;
    if (i < n4) p[i] = make_float4(0.f, 0.f, 0.f, 0.f);
}

// ---------------------------------------------------------------------------
// Fused triple GEMM: reads the 16-row x stripe once from HBM, produces
//   out[:,0:128] = x@W0+b0   (stride 384)
//   y1           = x@W1+b1   (stride 128, workspace)
//   t2           = x@W2+b2   (stride 128, workspace)
// Block = 256 thr = 8 waves. 24 16x16 tiles per stripe -> 3 tiles per wave.
// Each tile: K=128 -> 32x v_wmma_f32_16x16x4_f32.
// ---------------------------------------------------------------------------
__global__ __launch_bounds__(256) void mh_gemm3(
    const float* __restrict__ x,
    const float* __restrict__ W0, const float* __restrict__ b0,
    const float* __restrict__ W1, const float* __restrict__ b1,
    const float* __restrict__ W2, const float* __restrict__ b2,
    float* __restrict__ out, float* __restrict__ y1, float* __restrict__ t2,
    int N)
{
    __shared__ float xs[16][IN_C + 4];   // +4 pad: LDS bank-conflict free A reads

    const int tid  = threadIdx.x;
    const int row0 = blockIdx.x * 16;

    // Cooperative stripe load: 2048 floats / 256 threads = 8 each
    for (int i = tid; i < 16 * IN_C; i += 256) {
        const int r = i >> 7, c = i & 127;
        const int gr = row0 + r;
        xs[r][c] = (gr < N) ? x[(size_t)gr * IN_C + c] : 0.f;
    }
    __syncthreads();

    // Force the wave id scalar: makes tile/matrix selection + base pointers
    // live in SGPRs (saddr-form B loads, scalar-branch epilogue).
    const int wave  = __builtin_amdgcn_readfirstlane(tid >> 5);
    const int lane  = tid & 31;
    const int lhalf = lane >> 4;   // 0: K+0/1 rows, 1: K+2/3 rows
    const int lmod  = lane & 15;

    #pragma unroll
    for (int ti = 0; ti < 3; ++ti) {
        const int t   = wave * 3 + ti;
        const int mat = t >> 3;          // 0..2 : which weight matrix (scalar)
        const int nt  = t & 7;           // 0..7 : 16-col tile        (scalar)
        const int n0  = nt * 16;
        const float* __restrict__ W = (mat == 0) ? W0 : (mat == 1) ? W1 : W2;
        const float* __restrict__ b = (mat == 0) ? b0 : (mat == 1) ? b1 : b2;

        v8f acc = {};
        #pragma unroll 8
        for (int k0 = 0; k0 < IN_C; k0 += 4) {
            const int kk = k0 + 2 * lhalf;
            // A fragment (16x4): lane m holds x[m, kk] / x[m, kk+1]
            v2f a;
            a.x = xs[lmod][kk + 0];
            a.y = xs[lmod][kk + 1];
            // B fragment (4x16): lane n holds W[kk, n0+n] / W[kk+1, n0+n]
            v2f bb;
            bb.x = W[(size_t)(kk + 0) * OUT_C + n0 + lmod];
            bb.y = W[(size_t)(kk + 1) * OUT_C + n0 + lmod];
            acc = __builtin_amdgcn_wmma_f32_16x16x4_f32(
                      false, a, false, bb, (short)0, acc, false, false);
        }

        const float bias = b[n0 + lmod];
        float* __restrict__ dst;
        int stride;
        if (mat == 0)      { dst = out; stride = 3 * OUT_C; }
        else if (mat == 1) { dst = y1;  stride = OUT_C;     }
        else               { dst = t2;  stride = OUT_C;     }

        // C/D layout: VGPR r -> row M = r + 8*lhalf, col N = lmod
        #pragma unroll
        for (int r = 0; r < 8; ++r) {
            const int m = row0 + r + 8 * lhalf;
            if (m < N)
                dst[(size_t)m * stride + n0 + lmod] = acc[r] + bias;
        }
    }
}

// ---------------------------------------------------------------------------
// Edge-parallel SpMM: dst[row[e], :] += w[e] * src[col[e], :]
// One wave per edge; lane owns float4 (128 ch / 32 lanes). Gather B128
// (512 B/wave, fully coalesced), scatter with global_atomic_add_f32.
// ---------------------------------------------------------------------------
__global__ __launch_bounds__(256) void mh_spmm(
    const int* __restrict__ row, const int* __restrict__ col,
    const float* __restrict__ w,
    const float* __restrict__ src,     // [N,128]
    float* __restrict__ dst,           // base already offset to target cols
    int dstStride, long E)
{
    const int wave = threadIdx.x >> 5;
    const int lane = threadIdx.x & 31;
    const long e = (long)blockIdx.x * 8 + wave;
    if (e >= E) return;

    // Prefetch a future edge's source row into cache (gfx1250 global_prefetch)
    const long ep = e + 512;
    if (ep < E) {
        const int cp = col[ep];
        __builtin_prefetch(src + (size_t)cp * IN_C + lane * 4, 0, 0);
    }

    const int   r  = row[e];
    const int   c  = col[e];
    const float wt = w[e];

    const float4 v = *(const float4*)(src + (size_t)c * IN_C + lane * 4);

    float* d = dst + (size_t)r * dstStride + lane * 4;
    atomic_fadd_native(d + 0, wt * v.x);
    atomic_fadd_native(d + 1, wt * v.y);
    atomic_fadd_native(d + 2, wt * v.z);
    atomic_fadd_native(d + 3, wt * v.w);
}

// ---------------------------------------------------------------------------
// Launch: zero accumulators -> fused GEMM -> SpMM hop1 -> re-zero scratch ->
// SpMM hop2a -> SpMM hop2b. All on `stream`, graph-capture safe.
// ---------------------------------------------------------------------------
extern "C" void kernel_launch(void* const* d_in, const int* in_sizes, int n_in,
                              void* d_out, int out_size, void* d_ws, size_t ws_size,
                              hipStream_t stream) {
    const float* x  = (const float*)d_in[0];
    const int*   rw = (const int*)  d_in[1];
    const int*   cl = (const int*)  d_in[2];
    const float* ew = (const float*)d_in[3];
    const float* W0 = (const float*)d_in[4];
    const float* b0 = (const float*)d_in[5];
    const float* W1 = (const float*)d_in[6];
    const float* b1 = (const float*)d_in[7];
    const float* W2 = (const float*)d_in[8];
    const float* b2 = (const float*)d_in[9];

    const int  N = in_sizes[0] / IN_C;
    const long E = in_sizes[1];

    float* out  = (float*)d_out;                     // [N, 384]
    float* bufA = (float*)d_ws;                      // [N, 128]  y1, then s2
    float* bufB = bufA + (size_t)N * IN_C;           // [N, 128]  t2
    (void)ws_size; (void)n_in; (void)out_size;

    // 1) zero the output accumulator regions (full [N,384])
    {
        const long n4 = ((long)N * 3 * OUT_C) / 4;
        mh_zero4<<<(int)((n4 + 255) / 256), 256, 0, stream>>>((float4*)out, n4);
    }

    // 2) fused triple GEMM (WMMA f32)
    {
        const int nblocks = (N + 15) / 16;
        mh_gemm3<<<nblocks, 256, 0, stream>>>(x, W0, b0, W1, b1, W2, b2,
                                              out, bufA, bufB, N);
    }

    // 3) hop1: out[:,128:256] += spmm(y1)
    {
        const int nblocks = (int)((E + 7) / 8);
        mh_spmm<<<nblocks, 256, 0, stream>>>(rw, cl, ew, bufA,
                                             out + OUT_C, 3 * OUT_C, E);
    }

    // 4) re-zero bufA to serve as s2 accumulator
    {
        const long n4 = ((long)N * IN_C) / 4;
        mh_zero4<<<(int)((n4 + 255) / 256), 256, 0, stream>>>((float4*)bufA, n4);
    }

    // 5) hop2 first propagation: s2 = spmm(t2)
    {
        const int nblocks = (int)((E + 7) / 8);
        mh_spmm<<<nblocks, 256, 0, stream>>>(rw, cl, ew, bufB,
                                             bufA, OUT_C, E);
    }

    // 6) hop2 second propagation: out[:,256:384] += spmm(s2)
    {
        const int nblocks = (int)((E + 7) / 8);
        mh_spmm<<<nblocks, 256, 0, stream>>>(rw, cl, ew, bufA,
                                             out + 2 * OUT_C, 3 * OUT_C, E);
    }
}